// UnsupervisedLoss_10685878632585
// MI455X (gfx1250) — compile-verified
//
#include <hip/hip_runtime.h>

typedef __attribute__((ext_vector_type(2))) float v2f;
typedef __attribute__((ext_vector_type(4))) float v4f;
typedef __attribute__((ext_vector_type(8))) float v8f;

#define KNN_PEN 1.0f
#define OPP_PEN 1.0f
#define DYN_PEN 0.5f
#define STAT_PEN 0.5f
#define OCCL_PEN 0.1f
#define STATIC_FLOW_PEN 1.0f
#define TRAFO_PEN 1.0f

#define BATCH 2
#define NPTS 2048
#define NFLOW 3
#define NTILES (NPTS / 16)     // 128
#define WAVES_PER_BLK 8        // 256 threads
#define TILES_PER_BLK 8        // one row-tile per wave
#define BLK_GROUPS (NTILES / TILES_PER_BLK)  // 16

__global__ __launch_bounds__(32)
void ul_init_kernel(float* accum) {
    if (threadIdx.x < 8) accum[threadIdx.x] = 0.0f;
}

// Map float bits to an order-preserving unsigned key.
__device__ __forceinline__ unsigned int float_to_okey(float f) {
    unsigned int b = __float_as_uint(f);
    return (b & 0x80000000u) ? ~b : (b | 0x80000000u);
}
__device__ __forceinline__ float okey_to_float(unsigned int k) {
    unsigned int b = (k & 0x80000000u) ? (k ^ 0x80000000u) : ~k;
    return __uint_as_float(b);
}

// 8 waves per block; each wave owns one 16-row source tile of (dir, t, b).
// Target cloud staged once in LDS as {x, y, z, |q|^2} per point; the score
// matrix s(n,m) = -2*w(n).t(m) + |t(m)|^2 is built with V_WMMA_F32_16X16X4_F32.
__global__ __launch_bounds__(256)
void ul_knn_wmma_kernel(const float* __restrict__ pc0,
                        const float* __restrict__ pc1,
                        const float* __restrict__ flows_fw,
                        const float* __restrict__ flows_bw,
                        float* __restrict__ accum) {
    __shared__ float spts[NPTS * 4];          // 32 KB: [m] = {qx, qy, qz, q2}

    const int tid  = threadIdx.x;             // 0..255
    const int lane = tid & 31;                // 0..31 within wave
    const int wave = tid >> 5;                // 0..7

    int id = blockIdx.x;
    const int tgrp = id & (BLK_GROUPS - 1); id >>= 4;
    const int b    = id & 1;                id >>= 1;
    const int t    = id % NFLOW;            id /= NFLOW;
    const int dir  = id;                    // 0 = fw, 1 = bw

    const float* src_pc = dir ? pc1 : pc0;
    const float* tgt_pc = dir ? pc0 : pc1;
    const float* src_fl = dir ? flows_bw : flows_fw;
    const float* tgt_fl = dir ? flows_fw : flows_bw;

    // ---- Cooperative stage: target batch cloud -> LDS, with |q|^2 precomputed.
    const float* tb = tgt_pc + (size_t)b * NPTS * 3;
    for (int i = tid; i < NPTS; i += 256) {   // exactly 8 uniform iterations
        const float qx = tb[i * 3 + 0];
        const float qy = tb[i * 3 + 1];
        const float qz = tb[i * 3 + 2];
        v4f s;
        s.x = qx; s.y = qy; s.z = qz;
        s.w = qx * qx + qy * qy + qz * qz;
        *(v4f*)(spts + i * 4) = s;            // ds_store_b128
    }
    __syncthreads();

    // ---- Per-wave source tile setup.
    const int row  = lane & 15;
    const int half = lane >> 4;
    const int nTile = tgrp * TILES_PER_BLK + wave;
    const int n    = nTile * 16 + row;

    const int pbase = (b * NPTS + n) * 3;
    const float px = src_pc[pbase + 0];
    const float py = src_pc[pbase + 1];
    const float pz = src_pc[pbase + 2];
    const int fbase = ((t * BATCH + b) * NPTS + n) * 3;
    const float fx = src_fl[fbase + 0];
    const float fy = src_fl[fbase + 1];
    const float fz = src_fl[fbase + 2];
    const float wx = px + fx, wy = py + fy, wz = pz + fz;
    const float w2 = wx * wx + wy * wy + wz * wz;

    // A tile (16x4 f32): lanes 0-15 hold K0/K1 = {-2wx,-2wy}; lanes 16-31 K2/K3 = {-2wz, 1}
    v2f a;
    a.x = half ? (-2.0f * wz) : (-2.0f * wx);
    a.y = half ? 1.0f         : (-2.0f * wy);

    float rmin[8];
    int   rarg[8];
#pragma unroll
    for (int r = 0; r < 8; ++r) { rmin[r] = 3.4e38f; rarg[r] = 0; }

    // ---- Column loop, fed entirely from LDS.
#pragma unroll 2
    for (int mTile = 0; mTile < NTILES; ++mTile) {
        const int m = mTile * 16 + row;
        const v4f q = *(const v4f*)(spts + m * 4);   // ds_load_b128, conflict-free

        // B tile (4x16 f32): lane<16 -> {K0,K1} = {qx,qy}; lane>=16 -> {K2,K3} = {qz,|q|^2}
        v2f bm;
        bm.x = half ? q.z : q.x;
        bm.y = half ? q.w : q.y;

        v8f c = {};
        v8f d = __builtin_amdgcn_wmma_f32_16x16x4_f32(
            /*neg_a=*/false, a, /*neg_b=*/false, bm,
            /*c_mod=*/(short)0, c, /*reuse_a=*/false, /*reuse_b=*/false);

        // D layout: lane l, vgpr r -> row M = r + 8*(l>>4), col N = l&15 == this lane's m.
#pragma unroll
        for (int r = 0; r < 8; ++r) {
            const float dv = d[r];
            const bool lt = dv < rmin[r];
            rmin[r] = lt ? dv : rmin[r];
            rarg[r] = lt ? m : rarg[r];
        }
    }

    // Pack (value, index) into monotonic u64 keys: unsigned min == lexicographic
    // (min value, then min index) -> matches jnp.argmin first-occurrence.
    unsigned long long key[8];
#pragma unroll
    for (int r = 0; r < 8; ++r)
        key[r] = ((unsigned long long)float_to_okey(rmin[r]) << 32) | (unsigned int)rarg[r];

    // Branchless butterfly min within each 16-lane half.
#pragma unroll
    for (int mask = 1; mask <= 8; mask <<= 1) {
#pragma unroll
        for (int r = 0; r < 8; ++r) {
            const unsigned long long ok = __shfl_xor(key[r], mask, 32);
            key[r] = (ok < key[r]) ? ok : key[r];
        }
    }

    // Distribute: lane L (0..15) gets row L's key, via pure selects.
    unsigned long long myKey = 0ull;
#pragma unroll
    for (int r = 0; r < 8; ++r) {
        const unsigned long long klo = __shfl(key[r], 0, 32);   // rows 0..7
        const unsigned long long khi = __shfl(key[r], 16, 32);  // rows 8..15
        myKey = (lane == r)     ? klo : myKey;
        myKey = (lane == r + 8) ? khi : myKey;
    }

    float nd = 0.0f, op = 0.0f;
    if (lane < 16) {
        const int   myArg = (int)(unsigned int)myKey;
        const float myMin = okey_to_float((unsigned int)(myKey >> 32));
        nd = fmaxf(myMin + w2, 0.0f);            // recover clamped sq-distance
        if (t == 0) {                            // cycle loss only uses flow type 0
            const int gbase = (b * NPTS + myArg) * 3;  // tgt_fl[t=0]
            const float gx = tgt_fl[gbase + 0];
            const float gy = tgt_fl[gbase + 1];
            const float gz = tgt_fl[gbase + 2];
            const float ox = fx + gx, oy = fy + gy, oz = fz + gz;
            op = ox * ox + oy * oy + oz * oz;
        }
    }
#pragma unroll
    for (int mask = 16; mask >= 1; mask >>= 1) {
        nd += __shfl_xor(nd, mask, 32);
        op += __shfl_xor(op, mask, 32);
    }
    if (lane == 0) {
        atomicAdd(&accum[dir * 3 + t], nd);
        if (t == 0) atomicAdd(&accum[6 + dir], op);
    }
}

__global__ __launch_bounds__(256)
void ul_finalize_kernel(const float* __restrict__ pc0,
                        const float* __restrict__ pc1,
                        const float* __restrict__ flows_fw,
                        const float* __restrict__ flows_bw,
                        const float* __restrict__ disap_fw,
                        const float* __restrict__ disap_bw,
                        const float* __restrict__ trafo_fw,
                        const float* __restrict__ trafo_bw,
                        const float* __restrict__ stat_fw,
                        const float* __restrict__ stat_bw,
                        const float* __restrict__ accum,
                        float* __restrict__ out) {
    __shared__ float s_occ[256];
    __shared__ float s_sf[256];
    const int tid = threadIdx.x;
    float occ = 0.0f, sf = 0.0f;

    for (int j = tid; j < BATCH * NPTS; j += 256) {
        occ += disap_fw[j] + disap_bw[j];
        const int bb = j / NPTS;
        const int n  = j % NPTS;
        {
            const float* T = trafo_fw + bb * 16;
            const float* p = pc0 + j * 3;
            const float ix = T[0]*p[0] + T[1]*p[1] + T[2]*p[2]  + T[3]  - p[0];
            const float iy = T[4]*p[0] + T[5]*p[1] + T[6]*p[2]  + T[7]  - p[1];
            const float iz = T[8]*p[0] + T[9]*p[1] + T[10]*p[2] + T[11] - p[2];
            const float* f = flows_fw + ((2 * BATCH + bb) * NPTS + n) * 3;
            const float dx = f[0] - ix, dy = f[1] - iy, dz = f[2] - iz;
            sf += stat_fw[j] * (dx*dx + dy*dy + dz*dz);
        }
        {
            const float* T = trafo_bw + bb * 16;
            const float* p = pc1 + j * 3;
            const float ix = T[0]*p[0] + T[1]*p[1] + T[2]*p[2]  + T[3]  - p[0];
            const float iy = T[4]*p[0] + T[5]*p[1] + T[6]*p[2]  + T[7]  - p[1];
            const float iz = T[8]*p[0] + T[9]*p[1] + T[10]*p[2] + T[11] - p[2];
            const float* f = flows_bw + ((2 * BATCH + bb) * NPTS + n) * 3;
            const float dx = f[0] - ix, dy = f[1] - iy, dz = f[2] - iz;
            sf += stat_bw[j] * (dx*dx + dy*dy + dz*dz);
        }
    }
    s_occ[tid] = occ;
    s_sf[tid]  = sf;
    __syncthreads();
    for (int s = 128; s > 0; s >>= 1) {
        if (tid < s) { s_occ[tid] += s_occ[tid + s]; s_sf[tid] += s_sf[tid + s]; }
        __syncthreads();
    }

    if (tid == 0) {
        const float inv = 1.0f / (float)(BATCH * NPTS);
        const float occl_loss = 0.5f * s_occ[0] * inv;
        const float sf_loss   = 0.5f * s_sf[0]  * inv;
        float tl = 0.0f;
        for (int bb = 0; bb < BATCH; ++bb) {
            const float* Tf = trafo_fw + bb * 16;
            const float* Tb = trafo_bw + bb * 16;
            for (int i = 0; i < 4; ++i)
                for (int k = 0; k < 4; ++k) {
                    float v = 0.0f;
                    for (int j2 = 0; j2 < 4; ++j2) v += Tf[i*4 + j2] * Tb[j2*4 + k];
                    v -= (i == k) ? 1.0f : 0.0f;
                    tl += v * v;
                }
        }
        tl /= (float)BATCH;

        const float flow_loss = 0.5f * (accum[0] + accum[3]) * inv;
        const float dyn_loss  = 0.5f * (accum[1] + accum[4]) * inv;
        const float stat_loss = 0.5f * (accum[2] + accum[5]) * inv;
        const float opp_loss  = 0.5f * (accum[6] + accum[7]) * inv;

        out[0] = STATIC_FLOW_PEN * sf_loss
               + TRAFO_PEN * tl
               + OCCL_PEN * occl_loss
               + KNN_PEN * flow_loss
               + DYN_PEN * dyn_loss
               + STAT_PEN * stat_loss
               + OPP_PEN * opp_loss;
    }
}

extern "C" void kernel_launch(void* const* d_in, const int* in_sizes, int n_in,
                              void* d_out, int out_size, void* d_ws, size_t ws_size,
                              hipStream_t stream) {
    (void)in_sizes; (void)n_in; (void)out_size; (void)ws_size;
    const float* pc0      = (const float*)d_in[0];
    const float* pc1      = (const float*)d_in[1];
    const float* flows_fw = (const float*)d_in[2];
    const float* flows_bw = (const float*)d_in[3];
    const float* disap_fw = (const float*)d_in[4];
    const float* disap_bw = (const float*)d_in[5];
    const float* trafo_fw = (const float*)d_in[6];
    const float* trafo_bw = (const float*)d_in[7];
    const float* stat_fw  = (const float*)d_in[8];
    const float* stat_bw  = (const float*)d_in[9];
    float* accum = (float*)d_ws;
    float* out   = (float*)d_out;

    ul_init_kernel<<<1, 32, 0, stream>>>(accum);
    // 2 dirs * 3 flow types * 2 batches * 16 tile-groups = 192 blocks of 8 waves
    ul_knn_wmma_kernel<<<2 * NFLOW * BATCH * BLK_GROUPS, 256, 0, stream>>>(
        pc0, pc1, flows_fw, flows_bw, accum);
    ul_finalize_kernel<<<1, 256, 0, stream>>>(
        pc0, pc1, flows_fw, flows_bw, disap_fw, disap_bw,
        trafo_fw, trafo_bw, stat_fw, stat_bw, accum, out);
}